// HartleyMultiHeadAttention_22368189677793
// MI455X (gfx1250) — compile-verified
//
#include <hip/hip_runtime.h>
#include <math.h>

typedef float v2f __attribute__((ext_vector_type(2)));
typedef float v8f __attribute__((ext_vector_type(8)));

// ---------------------------------------------------------------------------
// Generic strided-batched GEMM on the CDNA5 fp32 matrix pipe:
//   C[b](M,N) (+)= alpha * A[b](M,K) * B[b](K,N)
// using V_WMMA_F32_16X16X4_F32. 128 threads = 4 waves per block; block tile
// 64(M) x 64(N); wave w owns columns [bn+16w, bn+16w+16) for all 4 M-subtiles.
// Fragment layouts per CDNA5 ISA 7.12.2 (wave32):
//   A 16x4 : lane = M%16 (+16 -> K+=2), vgpr v -> K = 2*(lane>>4)+v
//   B 4x16 : lane = N%16 (+16 -> K+=2), vgpr v -> K = 2*(lane>>4)+v
//   C 16x16: vgpr r, lanes 0-15 -> M=r, lanes 16-31 -> M=r+8, N=lane%16
// K must be a multiple of 4 (true for all call sites: 96,64,48,8192,216,128).
// Out-of-range M/N handled by multiplicative masks (EXEC stays all-ones).
// ---------------------------------------------------------------------------
__global__ __launch_bounds__(128) void gemm_f32_wmma(
    const float* __restrict__ A, long sAm, long sAk, long sAb,
    const float* __restrict__ B, long sBk, long sBn, long sBb,
    float* __restrict__ C, long sCm, long sCn, long sCb,
    int M, int N, int K, float alpha, int accum)
{
  const int lane = threadIdx.x & 31;
  const int wave = threadIdx.x >> 5;
  const int half = lane >> 4;
  const int l16  = lane & 15;
  const int mBase = blockIdx.x * 64;
  const int nBase = blockIdx.y * 64 + wave * 16;

  A += (long)blockIdx.z * sAb;
  B += (long)blockIdx.z * sBb;
  C += (long)blockIdx.z * sCb;

  v8f zero = {0.f,0.f,0.f,0.f,0.f,0.f,0.f,0.f};
  v8f acc[4];
#pragma unroll
  for (int t = 0; t < 4; ++t) acc[t] = zero;

  const float* aPtr[4];
  float aMask[4];
#pragma unroll
  for (int t = 0; t < 4; ++t) {
    int m = mBase + t * 16 + l16;
    bool v = (m < M);
    aMask[t] = v ? 1.f : 0.f;
    aPtr[t] = A + (long)(v ? m : 0) * sAm + (long)(2 * half) * sAk;
  }
  const int n = nBase + l16;
  const bool nv = (n < N);
  const float bMask = nv ? 1.f : 0.f;
  const float* bPtr = B + (long)(nv ? n : 0) * sBn + (long)(2 * half) * sBk;

  for (int k0 = 0; k0 < K; k0 += 4) {
    v2f bf;
    bf.x = bMask * bPtr[0];
    bf.y = bMask * bPtr[sBk];
    bPtr += 4 * sBk;
#pragma unroll
    for (int t = 0; t < 4; ++t) {
      v2f af;
      af.x = aMask[t] * aPtr[t][0];
      af.y = aMask[t] * aPtr[t][sAk];
      aPtr[t] += 4 * sAk;
      acc[t] = __builtin_amdgcn_wmma_f32_16x16x4_f32(
          false, af, false, bf, (short)0, acc[t], false, false);
    }
  }

  if (nv) {
#pragma unroll
    for (int t = 0; t < 4; ++t) {
#pragma unroll
      for (int r = 0; r < 8; ++r) {
        int m = mBase + t * 16 + r + 8 * half;
        if (m < M) {
          long idx = (long)m * sCm + (long)n * sCn;
          float val = alpha * acc[t][r];
          if (accum) C[idx] += val; else C[idx] = val;
        }
      }
    }
  }
}

// ---------------------------------------------------------------------------
// Twiddles. Forward F (48x96, row-major [k][n]): exp(-2*pi*i*f(k)*n/96),
// f(k)= k<24 ? k : k+48 (low + negative modes). Inverse V (96x48 [n][k]):
// exp(+2*pi*i*f(k)*n/96). Phase reduced mod 96 in integers for accuracy.
// ---------------------------------------------------------------------------
__global__ void gen_consts(float* Fre, float* Fim, float* Vre, float* Vim) {
  int i = blockIdx.x * 256 + threadIdx.x;
  if (i >= 48 * 96) return;
  int k = i / 96, nn = i % 96;
  int f = (k < 24) ? k : (k + 48);
  int ph = (f * nn) % 96;
  float th = 6.283185307179586f * (float)ph / 96.f;
  float c = cosf(th), s = sinf(th);
  Fre[i] = c;  Fim[i] = -s;
  Vre[nn * 48 + k] = c;  Vim[nn * 48 + k] = s;
}

__global__ void pack_wqkv(const float* __restrict__ wq, const float* __restrict__ wk,
                          const float* __restrict__ wv, float* __restrict__ dst) {
  int i = blockIdx.x * 256 + threadIdx.x;
  if (i >= 384 * 64) return;
  int ic = i % 64, r = i / 64;
  int t = r / 128, q = r % 128;               // q = h*16 + o
  const float* w = (t == 0) ? wq : ((t == 1) ? wk : wv);
  dst[i] = w[q * 64 + ic];
}

__global__ void addsub(const float* __restrict__ x, const float* __restrict__ y,
                       float* __restrict__ sum, float* __restrict__ dif, int nelem) {
  int i = blockIdx.x * 256 + threadIdx.x;
  if (i >= nelem) return;
  float a = x[i], b = y[i];
  sum[i] = a + b;
  dif[i] = a - b;
}

__global__ void selu_k(float* __restrict__ s, int nelem) {
  int i = blockIdx.x * 256 + threadIdx.x;
  if (i >= nelem) return;
  float x = s[i];
  const float lam = 1.0507009873554805f, alp = 1.6732632423543772f;
  s[i] = lam * (x > 0.f ? x : alp * (expf(x) - 1.f));
}

// (t,h,o,z,y,x) natural layout -> grouped (t,h, c'=(o,pz,py,px), tok=(nz,ny,nx))
__global__ void group_qkv(const float* __restrict__ src, float* __restrict__ dst) {
  int i = blockIdx.x * 256 + threadIdx.x;
  if (i >= 42467328) return;
  int s = i % 110592, r = i / 110592;
  int x = s % 48, y = (s / 48) % 48, z = s / 2304;
  int o = r % 16, h = (r / 16) % 8, t = r / 128;
  int px = x & 7, nx = x >> 3, py = y & 7, ny = y >> 3, pz = z & 7, nz = z >> 3;
  int cp = ((((o << 3) | pz) << 3) | py) << 3 | px;
  int tok = (nz * 6 + ny) * 6 + nx;
  dst[((long)((t * 8 + h) * 8192 + cp)) * 216 + tok] = src[i];
}

// grouped (h, c', tok) -> natural (ch=h*16+o, z,y,x)
__global__ void ungroup_o(const float* __restrict__ src, float* __restrict__ dst) {
  int i = blockIdx.x * 256 + threadIdx.x;
  if (i >= 14155776) return;
  int tok = i % 216, cp = (i / 216) % 8192, h = i / (216 * 8192);
  int px = cp & 7, py = (cp >> 3) & 7, pz = (cp >> 6) & 7, o = cp >> 9;
  int nx = tok % 6, ny = (tok / 6) % 6, nz = tok / 36;
  int z = nz * 8 + pz, y = ny * 8 + py, x = nx * 8 + px;
  dst[(long)(h * 16 + o) * 110592 + (long)(z * 48 + y) * 48 + x] = src[i];
}

// ---------------------------------------------------------------------------
extern "C" void kernel_launch(void* const* d_in, const int* in_sizes, int n_in,
                              void* d_out, int out_size, void* d_ws, size_t ws_size,
                              hipStream_t stream) {
  const float* in = (const float*)d_in[0];   // (1,64,96,96,96)
  const float* wq = (const float*)d_in[1];   // (8,16,64)
  const float* wk = (const float*)d_in[2];
  const float* wv = (const float*)d_in[3];
  const float* wo = (const float*)d_in[4];   // (16,128)
  float* out = (float*)d_out;                // (1,16,96,96,96)
  float* w = (float*)d_ws;

  // workspace arena (floats), lifetime-based aliasing; peak ~407 MB
  const long R1re_o = 0,          R1im_o = 28311552;
  const long R2re_o = 56623104,   R2im_o = 70778880;
  const long blk_o  = 84934656;
  const long QKV_o  = 0,          G_o    = 42467328;   // R1/R2 dead by then
  const long S_o    = 92012544,   Og_o   = 92385792;
  const long U_o    = 0,          spec_o = 14155776;   // QKV/G dead
  const long T1re_o = 15925248,   T1im_o = 19464192;
  const long T2re_o = 23003136,   T2im_o = 30081024;
  const long sumI_o = 37158912,   difI_o = 44236800;
  const long Wqkv_o = 106541568;
  const long Fre_o  = 106566144,  Fim_o  = 106570752;
  const long Vre_o  = 106575360,  Vim_o  = 106579968;
  const long WS_FLOATS = 106584576;                    // 426,338,304 bytes
  if (ws_size < (size_t)WS_FLOATS * sizeof(float)) return;

  float *R1re = w+R1re_o, *R1im = w+R1im_o, *R2re = w+R2re_o, *R2im = w+R2im_o;
  float *blk = w+blk_o, *QKV = w+QKV_o, *G = w+G_o, *S = w+S_o, *Og = w+Og_o;
  float *U = w+U_o, *spec = w+spec_o;
  float *T1re = w+T1re_o, *T1im = w+T1im_o, *T2re = w+T2re_o, *T2im = w+T2im_o;
  float *sumI = w+sumI_o, *difI = w+difI_o, *Wqkv = w+Wqkv_o;
  float *Fre = w+Fre_o, *Fim = w+Fim_o, *Vre = w+Vre_o, *Vim = w+Vim_o;

  auto gemm = [&](const float* A, long sAm, long sAk, long sAb,
                  const float* B, long sBk, long sBn, long sBb,
                  float* C, long sCm, long sCn, long sCb,
                  int M, int N, int K, int batch, float alpha, int accum) {
    dim3 g((unsigned)((M + 63) / 64), (unsigned)((N + 63) / 64), (unsigned)batch);
    gemm_f32_wmma<<<g, 128, 0, stream>>>(A, sAm, sAk, sAb, B, sBk, sBn, sBb,
                                         C, sCm, sCn, sCb, M, N, K, alpha, accum);
  };
  auto ew = [&](int n) { return dim3((unsigned)((n + 255) / 256)); };

  gen_consts<<<ew(4608), 256, 0, stream>>>(Fre, Fim, Vre, Vim);
  pack_wqkv<<<ew(24576), 256, 0, stream>>>(wq, wk, wv, Wqkv);

  // ---- forward truncated DFT (complex, separable), axis x then y then z ----
  // F1: (64*96*96 rows of 96) x F^T -> (.,48) complex
  gemm(in, 96, 1, 0, Fre, 1, 96, 0, R1re, 48, 1, 0, 589824, 48, 96, 1, 1.f, 0);
  gemm(in, 96, 1, 0, Fim, 1, 96, 0, R1im, 48, 1, 0, 589824, 48, 96, 1, 1.f, 0);
  // F2: batch (c,z)=6144: Y(48x48) = F(48x96) * X(96x48), complex
  gemm(Fre, 96, 1, 0, R1re, 48, 1, 4608, R2re, 48, 1, 2304, 48, 48, 96, 6144,  1.f, 0);
  gemm(Fim, 96, 1, 0, R1im, 48, 1, 4608, R2re, 48, 1, 2304, 48, 48, 96, 6144, -1.f, 1);
  gemm(Fre, 96, 1, 0, R1im, 48, 1, 4608, R2im, 48, 1, 2304, 48, 48, 96, 6144,  1.f, 0);
  gemm(Fim, 96, 1, 0, R1re, 48, 1, 4608, R2im, 48, 1, 2304, 48, 48, 96, 6144,  1.f, 1);
  // F3 + Hartley combine: blk = Fre*(Xre-Xim) - Fim*(Xre+Xim)  (batch c=64)
  addsub<<<ew(14155776), 256, 0, stream>>>(R2re, R2im, /*sum*/R1re, /*dif*/R1im, 14155776);
  gemm(Fre, 96, 1, 0, R1im, 2304, 1, 221184, blk, 2304, 1, 110592, 48, 2304, 96, 64,  1.f, 0);
  gemm(Fim, 96, 1, 0, R1re, 2304, 1, 221184, blk, 2304, 1, 110592, 48, 2304, 96, 64, -1.f, 1);

  // ---- QKV projection: (384x64) * (64 x 110592) ----
  gemm(Wqkv, 64, 1, 0, blk, 110592, 1, 0, QKV, 110592, 1, 0, 384, 110592, 64, 1, 1.f, 0);
  group_qkv<<<ew(42467328), 256, 0, stream>>>(QKV, G);

  // ---- attention, per head (batch 8) ----
  const long HD = 8192L * 216L;  // per-head grouped size
  // S = selu( (1/sqrt(8192)) * Q^T K ),  M=N=216, K=8192
  gemm(G, 1, 216, HD, G + 8 * HD, 216, 1, HD, S, 216, 1, 46656,
       216, 216, 8192, 8, 0.011048543456039806f, 0);
  selu_k<<<ew(373248), 256, 0, stream>>>(S, 373248);
  // O[c,m] = sum_n V[c,n] S[m,n] : M=8192, N=216, K=216
  gemm(G + 16 * HD, 216, 1, HD, S, 1, 216, 46656, Og, 216, 1, HD,
       8192, 216, 216, 8, 1.f, 0);
  ungroup_o<<<ew(14155776), 256, 0, stream>>>(Og, U);

  // ---- output projection: (16x128) * (128 x 110592) ----
  gemm(wo, 128, 1, 0, U, 110592, 1, 0, spec, 110592, 1, 0, 16, 110592, 128, 1, 1.f, 0);

  // ---- inverse: put_corners + idht == separable inverse DFT, (Re+Im)/N ----
  // I1: (16*48*48 rows of 48) x V^T -> (.,96) complex
  gemm(spec, 48, 1, 0, Vre, 1, 48, 0, T1re, 96, 1, 0, 36864, 96, 48, 1, 1.f, 0);
  gemm(spec, 48, 1, 0, Vim, 1, 48, 0, T1im, 96, 1, 0, 36864, 96, 48, 1, 1.f, 0);
  // I2: batch (c,kz)=768: Y(96x96) = V(96x48) * X(48x96), complex
  gemm(Vre, 48, 1, 0, T1re, 96, 1, 4608, T2re, 96, 1, 9216, 96, 96, 48, 768,  1.f, 0);
  gemm(Vim, 48, 1, 0, T1im, 96, 1, 4608, T2re, 96, 1, 9216, 96, 96, 48, 768, -1.f, 1);
  gemm(Vre, 48, 1, 0, T1im, 96, 1, 4608, T2im, 96, 1, 9216, 96, 96, 48, 768,  1.f, 0);
  gemm(Vim, 48, 1, 0, T1re, 96, 1, 4608, T2im, 96, 1, 9216, 96, 96, 48, 768,  1.f, 1);
  // I3 + combine: out = ( Vre*(Xre+Xim) + Vim*(Xre-Xim) ) / N   (batch c=16)
  addsub<<<ew(7077888), 256, 0, stream>>>(T2re, T2im, sumI, difI, 7077888);
  const float invN = 1.f / 884736.f;
  gemm(Vre, 48, 1, 0, sumI, 9216, 1, 442368, out, 9216, 1, 884736, 96, 9216, 48, 16, invN, 0);
  gemm(Vim, 48, 1, 0, difI, 9216, 1, 442368, out, 9216, 1, 884736, 96, 9216, 48, 16, invN, 1);
}